// MemoryDecoder_88862873354548
// MI455X (gfx1250) — compile-verified
//
#include <hip/hip_runtime.h>
#include <hip/hip_bf16.h>

// ---------------- problem constants (from the reference) ----------------
#define Bz   2
#define H1z  64
#define W1z  96
#define H2z  64
#define W2z  96
#define NPOS (Bz * H1z * W1z)          // 12288
#define HWz  (H1z * W1z)               // 6144 (also H2*W2)
#define CORR_ELEMS (Bz * 81 * HWz)     // 995328
#define UP_ELEMS   (Bz * 2 * 8 * H1z * 8 * W1z) // 1572864

typedef __attribute__((ext_vector_type(2))) float  v2f;
typedef __attribute__((ext_vector_type(8))) float  v8f;
typedef __attribute__((ext_vector_type(4))) unsigned int u32x4;
typedef __attribute__((ext_vector_type(4))) int    i32x4;
typedef __attribute__((ext_vector_type(8))) int    i32x8;

// =======================================================================
// Kernel 1: corr = 81-point bilinear lookup into per-position cost maps.
// Thread mapping: gid -> (k, pos) with pos minor => coalesced coords reads
// and coalesced output writes; cost-map gathers are L2-resident (windows
// reused 81x across k).
// =======================================================================
__global__ __launch_bounds__(256)
void corr_lookup_kernel(const float* __restrict__ cost_maps,
                        const float* __restrict__ coords,
                        float* __restrict__ out)
{
    int gid = blockIdx.x * blockDim.x + threadIdx.x;
    if (gid >= 81 * NPOS) return;
    int k   = gid / NPOS;          // 0..80  (k = i*9 + j; x gets i-4, y gets j-4)
    int pos = gid - k * NPOS;      // 0..12287, == (b*H1 + h)*W1 + w
    int b   = pos / HWz;
    int hw  = pos - b * HWz;

    float cx = coords[(b * 2 + 0) * HWz + hw];
    float cy = coords[(b * 2 + 1) * HWz + hw];
    float x = cx + (float)(k / 9 - 4);
    float y = cy + (float)(k % 9 - 4);

    float x0f = floorf(x), y0f = floorf(y);
    float wx = x - x0f, wy = y - y0f;
    int ix = (int)x0f, iy = (int)y0f;

    const float* base = cost_maps + (size_t)pos * HWz;
    auto g = [&](int xx, int yy) -> float {
        if (xx < 0 || xx > W2z - 1 || yy < 0 || yy > H2z - 1) return 0.0f;
        return base[yy * W2z + xx];
    };
    float v = g(ix,     iy    ) * (1.0f - wx) * (1.0f - wy)
            + g(ix + 1, iy    ) * wx          * (1.0f - wy)
            + g(ix,     iy + 1) * (1.0f - wx) * wy
            + g(ix + 1, iy + 1) * wx          * wy;

    out[(size_t)(b * 81 + k) * HWz + hw] = v;
}

// =======================================================================
// Kernel 2: convex upsample.
// Block = 128 threads (4 waves), tile = 8 consecutive w-positions.
//  1. TDM DMA of the strided mask tile (576 x 8, row stride 6144 floats)
//     into LDS, synced with s_wait_tensorcnt.
//  2. Stage 8*flow 3x3 patches into LDS: K padded 9->12, N padded 2->16
//     with zeros so WMMA B operands are unconditional ds_loads.
//  3. Softmax over the 9 taps per (subpixel, position).
//  4. Per position: D[64x2] = W[64x12] @ P[12x2] via v_wmma_f32_16x16x4_f32
//     (4 M-tiles x 3 K-chunks, N=16 with 2 live columns).
// =======================================================================
struct UpSmem {
    float raw[576][8];     // mask tile  (must stay first: TDM lds_addr = 0)
    float Wl[8][64][12];   // softmax weights, K padded to 12 with zeros
    float Pl[8][12][16];   // 8*flow patches; cols 2..15 and k 9..11 are zero
};

__global__ __launch_bounds__(128)
void upsample_flow_kernel(const float* __restrict__ flow,
                          const float* __restrict__ mask,
                          float* __restrict__ out_up)
{
    __shared__ UpSmem s;

    const int bw  = W1z / 8;                 // 12 tiles per row
    int n   = blockIdx.x / (H1z * bw);
    int rem = blockIdx.x % (H1z * bw);
    int h   = rem / bw;
    int w0  = (rem % bw) * 8;

    int tid = threadIdx.x;

    // ---- stage 1: mask tile -> LDS via Tensor Data Mover --------------
#if __has_builtin(__builtin_amdgcn_tensor_load_to_lds) && __has_builtin(__builtin_amdgcn_s_wait_tensorcnt)
    const bool use_tdm = true;
    if (tid < 32) {
        unsigned long long ga =
            (unsigned long long)(const void*)(mask + (size_t)(n * 576) * HWz + h * W1z + w0);
        u32x4 g0;
        g0.x = 1u;                                   // count=1, user descriptor
        g0.y = 0u;                                   // lds_addr: s.raw at offset 0
        g0.z = (unsigned)(ga & 0xFFFFFFFFu);         // global_addr[31:0]
        g0.w = (unsigned)((ga >> 32) & 0x1FFFFFFu)   // global_addr[56:32]
             | (2u << 30);                           // type = 2 ("image")
        i32x8 g1;
        g1[0] = (int)(2u << 16);     // data_size = 2 (4 bytes/elem)
        g1[1] = (int)(8u << 16);     // tensor_dim0[15:0] = 8
        g1[2] = (int)(576u << 16);   // tensor_dim1[15:0] = 576
        g1[3] = (int)(8u << 16);     // tile_dim0 = 8
        g1[4] = 576;                 // tile_dim1 = 576, tile_dim2 = 0
        g1[5] = 6144;                // tensor_dim0_stride[31:0] = 6144 elems
        g1[6] = 0;
        g1[7] = 0;
        i32x4 gz = {0, 0, 0, 0};
        i32x8 gz8 = {0, 0, 0, 0, 0, 0, 0, 0};
        __builtin_amdgcn_tensor_load_to_lds(g0, g1, gz, gz, gz8, 0);
    }
#else
    const bool use_tdm = false;
    for (int i = tid; i < 576 * 8; i += 128) {
        int ch = i >> 3, p = i & 7;
        s.raw[ch][p] = mask[(size_t)(n * 576 + ch) * HWz + h * W1z + w0 + p];
    }
#endif

    // ---- stage 2: flow patches -> LDS (overlaps the DMA) --------------
    // Pl[p][k][c]: live data only at k<9, c<2; everything else zero so the
    // WMMA B operand is a plain unpredicated ds_load.
    for (int i = tid; i < 8 * 12 * 16; i += 128) {
        int p = i / 192;
        int r = i % 192;
        int k = r >> 4;
        int c = r & 15;
        float val = 0.0f;
        if (k < 9 && c < 2) {
            int ki = k / 3, kj = k % 3;
            int rr = h - 1 + ki;
            int cc = w0 + p - 1 + kj;
            if (rr >= 0 && rr < H1z && cc >= 0 && cc < W1z)
                val = 8.0f * flow[((size_t)(n * 2 + c) * H1z + rr) * W1z + cc];
        }
        s.Pl[p][k][c] = val;
    }

#if __has_builtin(__builtin_amdgcn_tensor_load_to_lds) && __has_builtin(__builtin_amdgcn_s_wait_tensorcnt)
    if (use_tdm && tid < 32) __builtin_amdgcn_s_wait_tensorcnt(0);
#endif
    __syncthreads();

    // ---- stage 3: softmax over 9 taps per (subpixel, position) --------
    for (int i = tid; i < 8 * 64; i += 128) {
        int p = i & 7;
        int sp = i >> 3;                 // subpixel 0..63
        float v[9];
        float mx = -3.4e38f;
        for (int t = 0; t < 9; ++t) {
            v[t] = s.raw[t * 64 + sp][p];
            mx = fmaxf(mx, v[t]);
        }
        float sum = 0.0f;
        for (int t = 0; t < 9; ++t) { v[t] = expf(v[t] - mx); sum += v[t]; }
        float inv = 1.0f / sum;
        for (int t = 0; t < 9; ++t)  s.Wl[p][sp][t] = v[t] * inv;
        s.Wl[p][sp][9] = 0.0f; s.Wl[p][sp][10] = 0.0f; s.Wl[p][sp][11] = 0.0f;
    }
    __syncthreads();

    // ---- stage 4: per-position [64x12] @ [12x2] -----------------------
#if __has_builtin(__builtin_amdgcn_wmma_f32_16x16x4_f32)
    {
        int wv   = tid >> 5;             // wave 0..3
        int lane = tid & 31;
        int l16  = lane & 15;
        int hi   = lane >> 4;            // A/B: K offset +2 for lanes 16..31
        for (int pp = 0; pp < 2; ++pp) {
            int p = wv + pp * 4;         // position within tile
            for (int mt = 0; mt < 4; ++mt) {
                v8f acc = {};
                for (int q = 0; q < 3; ++q) {
                    int k0  = q * 4 + hi * 2;
                    int row = mt * 16 + l16;
                    v2f a; a.x = s.Wl[p][row][k0];
                           a.y = s.Wl[p][row][k0 + 1];
                    v2f b; b.x = s.Pl[p][k0][l16];
                           b.y = s.Pl[p][k0 + 1][l16];
                    acc = __builtin_amdgcn_wmma_f32_16x16x4_f32(
                              false, a, false, b, (short)0, acc, false, false);
                }
                if (l16 < 2) {
                    int c = l16;
                    for (int i = 0; i < 8; ++i) {
                        int sp = mt * 16 + hi * 8 + i;   // D: vgpr i -> M=i / M=8+i
                        int i8 = sp >> 3, j8 = sp & 7;
                        out_up[(((size_t)(n * 2 + c) * (8 * H1z)) + (8 * h + i8)) * (8 * W1z)
                               + 8 * (w0 + p) + j8] = acc[i];
                    }
                }
            }
        }
    }
#else
    for (int i = tid; i < 8 * 64 * 2; i += 128) {
        int p = i & 7;
        int rest = i >> 3;
        int sp = rest & 63;
        int c  = rest >> 6;
        float acc = 0.0f;
        for (int t = 0; t < 9; ++t) acc += s.Wl[p][sp][t] * s.Pl[p][t][c];
        int i8 = sp >> 3, j8 = sp & 7;
        out_up[(((size_t)(n * 2 + c) * (8 * H1z)) + (8 * h + i8)) * (8 * W1z)
               + 8 * (w0 + p) + j8] = acc;
    }
#endif
}

// =======================================================================
extern "C" void kernel_launch(void* const* d_in, const int* in_sizes, int n_in,
                              void* d_out, int out_size, void* d_ws, size_t ws_size,
                              hipStream_t stream) {
    const float* cost_maps = (const float*)d_in[0];
    const float* coords    = (const float*)d_in[1];
    const float* flow      = (const float*)d_in[2];
    const float* mask      = (const float*)d_in[3];
    float* out = (float*)d_out;

    (void)in_sizes; (void)n_in; (void)out_size; (void)d_ws; (void)ws_size;

    // corr: 81 * 12288 threads, pos-minor
    int total = 81 * NPOS;
    corr_lookup_kernel<<<(total + 255) / 256, 256, 0, stream>>>(cost_maps, coords, out);

    // upsample: 1 block per 8-position tile
    int nblocks = Bz * H1z * (W1z / 8);   // 1536
    upsample_flow_kernel<<<nblocks, 128, 0, stream>>>(flow, mask, out + CORR_ELEMS);
}